// BasedKernel_82403242541263
// MI455X (gfx1250) — compile-verified
//
#include <hip/hip_runtime.h>

// Based-style feature map:
//   out[token] = [1, x/sqrt(sqrt(d)), (x (x) x)/(sqrt(2)*sqrt(d))],  d = 16
// x: [B,H,S,16] f32  ->  out: [B,H,S,273] f32
//
// Per token the 16x16 outer product is one V_WMMA_F32_16X16X4_F32:
//   A[16x4] = x*s2 in K-column 0 (cols 1..3 zero)
//   B[4x16] = x    in K-row 0    (rows 1..3 zero)
//   D = (x x^T) * s2, already in the canonical 16x16 f32 C/D lane layout.
//
// Memory-bound (~286 MiB streamed out > 192 MB L2): all traffic non-temporal.
// Fast path prefetches all TOK_PER_WAVE feature vectors in one load clause,
// then runs WMMA + 9-store clauses back to back (no per-token branches).

typedef __attribute__((ext_vector_type(2))) float v2f;
typedef __attribute__((ext_vector_type(8))) float v8f;

#define FEAT 16
#define OUTD 273            // 1 + 16 + 256
#define TOK_PER_WAVE 4
#define WAVES_PER_BLOCK 8   // 256 threads

__device__ __forceinline__ void emit_token(float val, float* __restrict__ o,
                                           int col, bool lo)
{
    const float INV_RRD  = 0.5f;                  // 1/sqrt(sqrt(16))
    const float INV_R2RD = 0.17677669529663689f;  // 1/(sqrt(2)*4)

    // A 16x4 f32: lane<16 -> (M=lane,K=0) = val*s2; K=1..3 zero
    v2f a; a.x = lo ? (val * INV_R2RD) : 0.0f; a.y = 0.0f;
    // B 4x16 f32: lane<16 -> (K=0,N=lane) = val; rows K=1..3 zero
    v2f b; b.x = lo ? val : 0.0f;              b.y = 0.0f;
    v8f c = {};

    // D = A x B + 0 = outer(x,x) * INV_R2RD  (f32, RNE)
    v8f d = __builtin_amdgcn_wmma_f32_16x16x4_f32(
        false, a, false, b, (short)0, c, false, false);

    // Branch-free linear part:
    //  lanes 0..15  -> o[1+col] = val/rrd
    //  lanes 16..31 -> o[0]     = 1.0   (same dword, same value: benign)
    const float lv = lo ? (val * INV_RRD) : 1.0f;
    const int   li = lo ? (1 + col) : 0;
    __builtin_nontemporal_store(lv, o + li);

    // Quadratic part: VGPR r = row r (lanes 0..15) / row r+8 (lanes 16..31)
    float* q = o + 17 + (lo ? 0 : 8) * 16 + col;
#pragma unroll
    for (int r = 0; r < 8; ++r) {
        __builtin_nontemporal_store(d[r], q + r * 16);
    }
}

__global__ __launch_bounds__(256) void based_featmap_wmma(
    const float* __restrict__ x, float* __restrict__ out, int n_tokens)
{
    const int lane = threadIdx.x & 31;
    const int col  = lane & 15;
    const bool lo  = lane < 16;

    const long wave = (long)blockIdx.x * WAVES_PER_BLOCK + (threadIdx.x >> 5);
    const long tok0 = wave * TOK_PER_WAVE;

    if (tok0 + TOK_PER_WAVE <= (long)n_tokens) {
        // ---- fast path: clause of 4 NT loads, one wait, then 4 WMMA+stores
        float val[TOK_PER_WAVE];
#pragma unroll
        for (int t = 0; t < TOK_PER_WAVE; ++t) {
            val[t] = __builtin_nontemporal_load(x + (tok0 + t) * FEAT + col);
        }
#pragma unroll
        for (int t = 0; t < TOK_PER_WAVE; ++t) {
            emit_token(val[t], out + (tok0 + t) * OUTD, col, lo);
        }
    } else {
        // ---- guarded tail (never taken when n_tokens % TOK_PER_WAVE == 0)
#pragma unroll
        for (int t = 0; t < TOK_PER_WAVE; ++t) {
            const long token = tok0 + t;        // wave-uniform
            if (token >= (long)n_tokens) return;
            const float v =
                __builtin_nontemporal_load(x + token * FEAT + col);
            emit_token(v, out + token * OUTD, col, lo);
        }
    }
}

extern "C" void kernel_launch(void* const* d_in, const int* in_sizes, int n_in,
                              void* d_out, int out_size, void* d_ws, size_t ws_size,
                              hipStream_t stream) {
    const float* x = (const float*)d_in[0];
    float* out = (float*)d_out;

    const int n_tokens = in_sizes[0] / FEAT;            // B*H*S
    const long waves   = ((long)n_tokens + TOK_PER_WAVE - 1) / TOK_PER_WAVE;
    const long blocks  = (waves + WAVES_PER_BLOCK - 1) / WAVES_PER_BLOCK;

    based_featmap_wmma<<<(int)blocks, 256, 0, stream>>>(x, out, n_tokens);
}